// TimeNormalization_4501125726545
// MI455X (gfx1250) — compile-verified
//
#include <hip/hip_runtime.h>

typedef __attribute__((ext_vector_type(2))) float v2f;
typedef __attribute__((ext_vector_type(8))) float v8f;
typedef __attribute__((ext_vector_type(4))) int   v4i;

// pointer-to-AS1 / AS3 vector types for the async global->LDS builtin
typedef __attribute__((address_space(1))) v4i* gv4i_p;
typedef __attribute__((address_space(3))) v4i* lv4i_p;

namespace {
constexpr int   kB     = 16;
constexpr int   kT     = 4096;
constexpr int   kF     = 512;
constexpr float kAlpha = 0.1f;
constexpr float kC     = 0.9f;    // 1 - alpha
constexpr float kEps   = 1e-3f;
constexpr int   kLStr  = 20;      // LDS row stride in floats (16 data + 4 pad, bank-conflict free)
constexpr int   kTiles = kT / 16; // 256 sequential 16-step tiles per wave
}

// CDNA5 async global->LDS path if the toolchain exposes the builtins; else ds_store fallback.
#if __has_builtin(__builtin_amdgcn_global_load_async_to_lds_b128) && \
    __has_builtin(__builtin_amdgcn_s_wait_asynccnt)
#define TN_ASYNC 1
#define TN_WAIT(n) __builtin_amdgcn_s_wait_asynccnt(n)
#else
#define TN_ASYNC 0
#define TN_WAIT(n)
#endif

// One wave per (batch b, 16-feature chunk). The wave walks T sequentially in 16-step
// tiles; each tile is computed as D = A(16x16 lower-tri decay) x P(16x16) via
// 4 chained v_wmma_f32_16x16x4_f32 per state (mean, mean-square).
__global__ __launch_bounds__(32)
void tn_ema_wmma_kernel(const float* __restrict__ x,
                        const float* __restrict__ st,
                        float* __restrict__ out)
{
  const int lane   = threadIdx.x;   // 0..31 (wave32)
  const int laneLo = lane & 15;     // N (feature) index within tile
  const int laneHi = lane >> 4;     // lane-half selector
  const int f0     = blockIdx.x * 16;
  const int b      = blockIdx.y;

  __shared__ float lds[2][16 * kLStr];

  // ---- constant decay matrix A[m][k] = alpha * C^(m-k) (k<=m), split into 4 K-slices ----
  // f32 A-operand layout (ISA 7.12.2): lanes 0-15 hold M=lane; VGPR v holds K' = v + 2*laneHi.
  v2f A[4];
#pragma unroll
  for (int kc = 0; kc < 4; ++kc) {
#pragma unroll
    for (int v = 0; v < 2; ++v) {
      const int k = 4 * kc + v + 2 * laneHi;
      const int m = laneLo;
      float val = 0.0f;
      if (m >= k) {
        val = kAlpha;
        for (int i = 0; i < m - k; ++i) val *= kC;
      }
      if (v == 0) A[kc].x = val; else A[kc].y = val;
    }
  }

  // carry decay per D row: cdec[r] = C^(M+1), with D layout M = r + 8*laneHi
  float cdec[8];
  {
    float p = kC;
    for (int i = 0; i < 8 * laneHi; ++i) p *= kC;
#pragma unroll
    for (int r = 0; r < 8; ++r) { cdec[r] = p; p *= kC; }
  }

  // initial state from input: state[0]=running mean, state[1]=running mean-square
  float mCar = st[(size_t)b * kF + f0 + laneLo];
  float qCar = st[(size_t)kB * kF + (size_t)b * kF + f0 + laneLo];

  const float* xb = x   + ((size_t)b * kT) * kF + f0;
  float*       ob = out + ((size_t)b * kT) * kF + f0;

  // Stage a 16(time) x 16(feat) fp32 tile into LDS: 2 x B128 per tile
  // (lane -> row = j*8 + lane/4, 16B chunk = lane%4; rows padded to 80B, 16B aligned).
  auto issue_copy = [&](int buf, int t0) {
    const int row = lane >> 2;
    const int chk = (lane & 3) * 4;  // float offset within row
#pragma unroll
    for (int j = 0; j < 2; ++j) {
      const int r = row + 8 * j;
      const float* g = xb + (size_t)(t0 + r) * kF + chk;
      float* l = &lds[buf][r * kLStr + chk];
#if TN_ASYNC
      __builtin_amdgcn_global_load_async_to_lds_b128((gv4i_p)g, (lv4i_p)l, 0, 0);
#else
      const float4 t = *(const float4*)g;   // fallback: global_load_b128 + ds_store_b128
      *(float4*)l = t;
#endif
    }
  };

  issue_copy(0, 0);  // prologue: tile 0 into buffer 0

#pragma unroll 1
  for (int tile = 0; tile < kTiles; ++tile) {
    const int cur = tile & 1;
    if (tile + 1 < kTiles) {
      issue_copy(cur ^ 1, (tile + 1) * 16);  // prefetch next tile into other buffer
      TN_WAIT(2);                            // current tile's 2 async ops complete
    } else {
      TN_WAIT(0);
    }

    // B-operand layout (mirrors f32 A): VGPR v holds K' = v + 2*laneHi, N = laneLo.
    v8f Dm = {0.f,0.f,0.f,0.f,0.f,0.f,0.f,0.f};
    v8f Dq = {0.f,0.f,0.f,0.f,0.f,0.f,0.f,0.f};
#pragma unroll
    for (int kc = 0; kc < 4; ++kc) {
      const int k0 = 4 * kc + 2 * laneHi;
      const float x0 = lds[cur][(k0 + 0) * kLStr + laneLo];
      const float x1 = lds[cur][(k0 + 1) * kLStr + laneLo];
      v2f Bm; Bm.x = x0;      Bm.y = x1;       // mean path: P = x
      v2f Bq; Bq.x = x0 * x0; Bq.y = x1 * x1;  // mean-square path: P = x^2
      Dm = __builtin_amdgcn_wmma_f32_16x16x4_f32(false, A[kc], false, Bm, (short)0, Dm, false, false);
      Dq = __builtin_amdgcn_wmma_f32_16x16x4_f32(false, A[kc], false, Bq, (short)0, Dq, false, false);
    }

    // Epilogue in D layout (M = r + 8*laneHi, N = laneLo): add carry term, normalize, store.
    float mLast = 0.f, qLast = 0.f;
#pragma unroll
    for (int r = 0; r < 8; ++r) {
      const int   M  = r + 8 * laneHi;
      const float xv = lds[cur][M * kLStr + laneLo];
      const float m  = Dm[r] + cdec[r] * mCar;
      const float q  = Dq[r] + cdec[r] * qCar;
      const float var = q - m * m;
      const float xn  = (xv - m) * rsqrtf(var + kEps);
      ob[(size_t)(tile * 16 + M) * kF + laneLo] = xn;
      if (r == 7) { mLast = m; qLast = q; }
    }
    // new carry = state at local t=15, held by lane (16 + N) in VGPR 7 -> broadcast
    mCar = __shfl(mLast, 16 + laneLo, 32);
    qCar = __shfl(qLast, 16 + laneLo, 32);
  }

  // final state [2, B, F] appended after x_norm
  if (lane < 16) {
    const size_t base = (size_t)kB * kT * kF;
    out[base + (size_t)b * kF + f0 + lane]                   = mCar;
    out[base + (size_t)kB * kF + (size_t)b * kF + f0 + lane] = qCar;
  }
}

extern "C" void kernel_launch(void* const* d_in, const int* in_sizes, int n_in,
                              void* d_out, int out_size, void* d_ws, size_t ws_size,
                              hipStream_t stream) {
  (void)in_sizes; (void)n_in; (void)out_size; (void)d_ws; (void)ws_size;
  const float* x  = (const float*)d_in[0];   // [16, 4096, 512] fp32
  const float* st = (const float*)d_in[1];   // [2, 16, 512] fp32 initial state
  float* out = (float*)d_out;                // [16,4096,512] x_norm ++ [2,16,512] final state
  dim3 grid(kF / 16, kB);                    // 32 feature-chunks x 16 batches = 512 waves
  tn_ema_wmma_kernel<<<grid, dim3(32), 0, stream>>>(x, st, out);
}